// ResidualS4Unit_19095424598419
// MI455X (gfx1250) — compile-verified
//
#include <hip/hip_runtime.h>

// Problem constants (match reference)
#define Bb   8
#define Hh   512
#define Ll   4096
#define Nn   32
#define LC   128           // chunk length
#define NCH  32            // chunks per sequence (Ll/LC)
#define COLS 256           // Bb*NCH columns per h in the chunk GEMM
#define KD   192           // LC + 2*Nn combined K dimension
#define BSTR 129           // padded LDS row stride for chunk GEMM B panel
#define VSTR 65            // padded LDS row stride for conv GEMM B panel

typedef float v2f __attribute__((ext_vector_type(2)));
typedef float v8f __attribute__((ext_vector_type(8)));

__device__ __forceinline__ v8f wmma4(v2f a, v2f b, v8f c) {
  // D(16x16,f32) = A(16x4,f32) x B(4x16,f32) + C
  return __builtin_amdgcn_wmma_f32_16x16x4_f32(
      /*neg_a=*/false, a, /*neg_b=*/false, b,
      /*c_mod=*/(short)0, c, /*reuse_a=*/false, /*reuse_b=*/false);
}

__device__ __forceinline__ float2 cmul(float2 a, float2 b) {
  return make_float2(a.x * b.x - a.y * b.y, a.x * b.y + a.y * b.x);
}
__device__ __forceinline__ float2 cpow_u(float2 w, unsigned e) {
  float2 r = make_float2(1.f, 0.f);
  float2 base = w;
  while (e) {
    if (e & 1u) r = cmul(r, base);
    base = cmul(base, base);
    e >>= 1;
  }
  return r;
}
__device__ __forceinline__ float snakef(float x, float a) {
  float s = __sinf(a * x);
  return x + s * s / (a + 1e-9f);
}

// ---------------------------------------------------------------------------
// K1: per (h,n) discretization: w = exp(dt*A), ctilde = C*(w-1)/A, wLc = w^LC
// ---------------------------------------------------------------------------
__global__ void k_precompute(const float* __restrict__ log_dt,
                             const float* __restrict__ A_real,
                             const float* __restrict__ A_imag,
                             const float* __restrict__ C_real,
                             const float* __restrict__ C_imag,
                             float2* __restrict__ wc, float2* __restrict__ wLc,
                             float2* __restrict__ ct) {
  int i = blockIdx.x * blockDim.x + threadIdx.x;
  if (i >= Hh * Nn) return;
  int h = i / Nn;
  float dt = __expf(log_dt[h]);
  float ar = A_real[i], ai = A_imag[i];
  float er = __expf(dt * ar);
  float2 w = make_float2(er * __cosf(dt * ai), er * __sinf(dt * ai));
  // ctilde = C * (w - 1) / A   (A undiscretized, ar = -0.5 so never zero)
  float2 num = make_float2(w.x - 1.f, w.y);
  float inv = 1.f / (ar * ar + ai * ai);
  float2 q = make_float2((num.x * ar + num.y * ai) * inv,
                         (num.y * ar - num.x * ai) * inv);
  ct[i] = cmul(make_float2(C_real[i], C_imag[i]), q);
  wc[i] = w;
  float2 p = w;
#pragma unroll
  for (int s = 0; s < 7; ++s) p = cmul(p, p);  // w^128
  wLc[i] = p;
}

// ---------------------------------------------------------------------------
// K2: per h tables: Toeplitz taps Ktap[l] = 2Re sum_n ct*w^l,
//     correction matrix PQ[l][n]=2Re(ct*w^{l+1}), PQ[l][32+n]=-2Im(ct*w^{l+1})
// ---------------------------------------------------------------------------
__global__ void k_tables(const float2* __restrict__ wc,
                         const float2* __restrict__ ct,
                         float* __restrict__ Ktap, float* __restrict__ PQ) {
  int h = blockIdx.x;
  int l = threadIdx.x;  // 128 threads
  float ksum = 0.f;
  for (int n = 0; n < Nn; ++n) {
    float2 w = wc[h * Nn + n];
    float2 c = ct[h * Nn + n];
    float2 t = cmul(c, cpow_u(w, (unsigned)l));  // ct * w^l
    ksum += 2.f * t.x;
    float2 t2 = cmul(t, w);  // ct * w^{l+1}
    PQ[(h * LC + l) * 64 + n] = 2.f * t2.x;
    PQ[(h * LC + l) * 64 + 32 + n] = -2.f * t2.y;
  }
  Ktap[h * LC + l] = ksum;
}

// ---------------------------------------------------------------------------
// K3: chunk summaries. One wave per (b,h,chunk); lane = mode n.
//     z = sum_t w^{LC-1-t} * snake(x[t])  via recurrence z <- w*z + u
// ---------------------------------------------------------------------------
__global__ void k_summary(const float* __restrict__ x,
                          const float* __restrict__ alpha1,
                          const float2* __restrict__ wc,
                          float2* __restrict__ zc) {
  int wid = blockIdx.x * 8 + (threadIdx.x >> 5);  // (b*Hh+h)*NCH + chunk
  int lane = threadIdx.x & 31;
  int chunk = wid & (NCH - 1);
  int h = (wid >> 5) & (Hh - 1);
  int b = wid >> 14;
  float a1 = alpha1[h];
  float2 w = wc[h * Nn + lane];
  const float* xp = x + ((size_t)(b * Hh + h)) * Ll + chunk * LC;
  float2 z = make_float2(0.f, 0.f);
  for (int t4 = 0; t4 < LC / 4; ++t4) {
    float4 xq = ((const float4*)xp)[t4];
#pragma unroll
    for (int s = 0; s < 4; ++s) {
      float xv = (s == 0) ? xq.x : (s == 1) ? xq.y : (s == 2) ? xq.z : xq.w;
      float u = snakef(xv, a1);
      z = make_float2(w.x * z.x - w.y * z.y + u, w.x * z.y + w.y * z.x);
    }
  }
  zc[(size_t)wid * Nn + lane] = z;
}

// ---------------------------------------------------------------------------
// K4: sequential chunk scan (32 steps). One wave per (b,h); lane = mode n.
//     Emits entering state per chunk into S[h][row][col], row = n / 32+n
//     (re/im), col = b*NCH + chunk  -> B-matrix rows 128..191 of chunk GEMM.
// ---------------------------------------------------------------------------
__global__ void k_scan(const float2* __restrict__ zc,
                       const float2* __restrict__ wLc,
                       float* __restrict__ S) {
  int wid = blockIdx.x * 8 + (threadIdx.x >> 5);  // b*Hh + h
  int lane = threadIdx.x & 31;
  int h = wid & (Hh - 1);
  int b = wid >> 9;
  float2 wl = wLc[h * Nn + lane];
  const float2* zp = zc + (size_t)wid * NCH * Nn + lane;
  float* Sre = S + ((size_t)h * 64 + lane) * COLS + b * NCH;
  float* Sim = S + ((size_t)h * 64 + 32 + lane) * COLS + b * NCH;
  float2 s = make_float2(0.f, 0.f);
  for (int j = 0; j < NCH; ++j) {
    Sre[j] = s.x;
    Sim[j] = s.y;
    float2 z = zp[(size_t)j * Nn];
    s = make_float2(wl.x * s.x - wl.y * s.y + z.x,
                    wl.x * s.y + wl.y * s.x + z.y);
  }
}

// ---------------------------------------------------------------------------
// K5: fused chunk GEMM per h (WMMA f32 16x16x4):
//   Y(128x256) = [ToeplitzLow(Ktap) | PQ](128x192) x [snake(U); S](192x256)
//   epilogue: + D*u, snake2 -> v (aliases d_out)
//   grid = (Hh, 2 column halves), block = 256 (8 waves), dyn LDS ~98 KB
//   Branch-free A-fragments: taps zero-padded so A[m][k] = tap_pad[128+m-k].
// ---------------------------------------------------------------------------
__global__ void k_chunk_gemm(const float* __restrict__ x,
                             const float* __restrict__ alpha1,
                             const float* __restrict__ alpha2,
                             const float* __restrict__ Dp,
                             const float* __restrict__ Ktap,
                             const float* __restrict__ PQ,
                             const float* __restrict__ S, float* v) {
  extern __shared__ float sm[];
  float* Bsm = sm;                  // KD x BSTR
  float* tap_pad = sm + KD * BSTR;  // 256: [0..127]=0, [128+d]=Ktap[d]
  int h = blockIdx.x;
  int c0 = blockIdx.y * 128;
  int tid = threadIdx.x, lane = tid & 31, wv = tid >> 5;
  float a1 = alpha1[h];

  // Stage zero-padded taps (all 256 threads write exactly one slot)
  tap_pad[tid] = (tid < LC) ? 0.f : Ktap[h * LC + (tid - LC)];
  // Stage U = snake(x) rows 0..127 (coalesced in t)
  for (int idx = tid; idx < LC * 128; idx += 256) {
    int c = idx >> 7, t = idx & 127;
    int colg = c0 + c, b = colg >> 5, ch = colg & 31;
    float xv = x[((size_t)(b * Hh + h)) * Ll + ch * LC + t];
    Bsm[t * BSTR + c] = snakef(xv, a1);
  }
  // Stage S rows 128..191
  for (int idx = tid; idx < 64 * 128; idx += 256) {
    int r = idx >> 7, c = idx & 127;
    Bsm[(LC + r) * BSTR + c] = S[((size_t)h * 64 + r) * COLS + c0 + c];
  }
  __syncthreads();

  int mlane = lane & 15;
  int mrow = wv * 16 + mlane;
  int khalf = (lane >> 4) << 1;  // 0 or 2
  const float* PQh = PQ + (size_t)h * LC * 64;
  v8f acc[8] = {};

  // --- Phase 1: lower-triangular Toeplitz region, k in [0,128) ---
  for (int kk = 0; kk < LC / 4; ++kk) {
    int k0 = kk * 4 + khalf;
    int tbase = 128 + mrow - k0;  // always in [2,255]
    v2f af;
    af[0] = tap_pad[tbase];
    af[1] = tap_pad[tbase - 1];
#pragma unroll
    for (int nt = 0; nt < 8; ++nt) {
      int nc = nt * 16 + mlane;
      v2f bf;
      bf[0] = Bsm[k0 * BSTR + nc];
      bf[1] = Bsm[(k0 + 1) * BSTR + nc];
      acc[nt] = wmma4(af, bf, acc[nt]);
    }
  }
  // --- Phase 2: state-correction region, k in [128,192) ---
  for (int kk = 0; kk < (KD - LC) / 4; ++kk) {
    int k0 = LC + kk * 4 + khalf;
    const float2 pq = *(const float2*)(PQh + mrow * 64 + (k0 - LC));
    v2f af;
    af[0] = pq.x;
    af[1] = pq.y;
#pragma unroll
    for (int nt = 0; nt < 8; ++nt) {
      int nc = nt * 16 + mlane;
      v2f bf;
      bf[0] = Bsm[k0 * BSTR + nc];
      bf[1] = Bsm[(k0 + 1) * BSTR + nc];
      acc[nt] = wmma4(af, bf, acc[nt]);
    }
  }

  float a2 = alpha2[h], Dh = Dp[h];
#pragma unroll
  for (int nt = 0; nt < 8; ++nt) {
    int nc = nt * 16 + mlane;
    int colg = c0 + nc, b = colg >> 5, ch = colg & 31;
    size_t obase = ((size_t)(b * Hh + h)) * Ll + ch * LC;
#pragma unroll
    for (int r = 0; r < 8; ++r) {
      int row = wv * 16 + r + ((lane >> 4) << 3);
      float u = Bsm[row * BSTR + nc];
      float y = acc[nt][r] + Dh * u;
      v[obase + row] = snakef(y, a2);
    }
  }
}

// ---------------------------------------------------------------------------
// K6: 1x1 conv GEMM (W 1024x512 times v 512xL per batch) + bias + GLU + residual
//     grid = (Bb, Ll/64), block = 256 (8 waves), dyn LDS = 512*65*4 = 130 KB.
//     v aliases out: each block consumes its (b, l-tile) panel into LDS first.
// ---------------------------------------------------------------------------
__global__ void k_conv_glu(const float* v, const float* __restrict__ x,
                           const float* __restrict__ W,
                           const float* __restrict__ bconv, float* out) {
  extern __shared__ float Vp[];  // 512 x VSTR
  int b = blockIdx.x;
  int l0 = blockIdx.y * 64;
  int tid = threadIdx.x, lane = tid & 31, wv = tid >> 5;
  for (int idx = tid; idx < Hh * 64; idx += 256) {
    int hh = idx >> 6, j = idx & 63;
    Vp[hh * VSTR + j] = v[((size_t)(b * Hh + hh)) * Ll + l0 + j];
  }
  __syncthreads();

  int mlane = lane & 15;
  int khalf = (lane >> 4) << 1;
  for (int pp = 0; pp < 4; ++pp) {
    int p = wv + pp * 8;  // channel pair tile 0..31
    const float* Wa = W + (size_t)(p * 16 + mlane) * Hh;
    const float* Wg = W + (size_t)(512 + p * 16 + mlane) * Hh;
    v8f aA[4] = {};
    v8f aG[4] = {};
    for (int kk = 0; kk < Hh / 4; ++kk) {
      int k0 = kk * 4 + khalf;
      const float2 wa2 = *(const float2*)(Wa + k0);
      const float2 wg2 = *(const float2*)(Wg + k0);
      v2f wa;
      wa[0] = wa2.x; wa[1] = wa2.y;
      v2f wg;
      wg[0] = wg2.x; wg[1] = wg2.y;
#pragma unroll
      for (int nt = 0; nt < 4; ++nt) {
        int j = nt * 16 + mlane;
        v2f bf;
        bf[0] = Vp[k0 * VSTR + j];
        bf[1] = Vp[(k0 + 1) * VSTR + j];
        aA[nt] = wmma4(wa, bf, aA[nt]);
        aG[nt] = wmma4(wg, bf, aG[nt]);
      }
    }
#pragma unroll
    for (int nt = 0; nt < 4; ++nt) {
      int l = l0 + nt * 16 + mlane;
#pragma unroll
      for (int r = 0; r < 8; ++r) {
        int m = r + ((lane >> 4) << 3);
        int ca = p * 16 + m;
        float av = aA[nt][r] + bconv[ca];
        float gv = aG[nt][r] + bconv[512 + ca];
        float yv = av * (1.f / (1.f + __expf(-gv)));
        size_t o = ((size_t)(b * Hh + ca)) * Ll + l;
        out[o] = x[o] + yv;
      }
    }
  }
}

// ---------------------------------------------------------------------------
extern "C" void kernel_launch(void* const* d_in, const int* in_sizes, int n_in,
                              void* d_out, int out_size, void* d_ws,
                              size_t ws_size, hipStream_t stream) {
  const float* x = (const float*)d_in[0];
  const float* alpha1 = (const float*)d_in[1];
  const float* log_dt = (const float*)d_in[2];
  const float* A_real = (const float*)d_in[3];
  const float* A_imag = (const float*)d_in[4];
  const float* C_real = (const float*)d_in[5];
  const float* C_imag = (const float*)d_in[6];
  const float* Dp = (const float*)d_in[7];
  const float* alpha2 = (const float*)d_in[8];
  const float* W = (const float*)d_in[9];
  const float* bconv = (const float*)d_in[10];
  float* out = (float*)d_out;
  float* ws = (float*)d_ws;

  // Workspace layout (floats)
  float2* wc = (float2*)(ws + 0);                 // 2*H*N
  float2* wLc = (float2*)(ws + 32768);            // 2*H*N
  float2* ct = (float2*)(ws + 65536);             // 2*H*N
  float* Ktap = ws + 98304;                       // H*LC
  float* PQ = ws + 163840;                        // H*LC*64
  float* S = ws + 4358144;                        // H*64*COLS
  float2* zc = (float2*)(ws + 12746752);          // B*H*NCH*N complex
  float* v = out;                                 // v aliases d_out

  k_precompute<<<(Hh * Nn + 255) / 256, 256, 0, stream>>>(
      log_dt, A_real, A_imag, C_real, C_imag, wc, wLc, ct);
  k_tables<<<Hh, LC, 0, stream>>>(wc, ct, Ktap, PQ);
  k_summary<<<Bb * Hh * NCH / 8, 256, 0, stream>>>(x, alpha1, wc, zc);
  k_scan<<<Bb * Hh / 8, 256, 0, stream>>>(zc, wLc, S);
  k_chunk_gemm<<<dim3(Hh, 2), 256, (KD * BSTR + 256) * sizeof(float), stream>>>(
      x, alpha1, alpha2, Dp, Ktap, PQ, S, v);
  k_conv_glu<<<dim3(Bb, Ll / 64), 256, Hh * VSTR * sizeof(float), stream>>>(
      v, x, W, bconv, out);
}